// FusedLocalNormalizedCrossCorrelationLoss_37838661877790
// MI455X (gfx1250) — compile-verified
//
#include <hip/hip_runtime.h>

typedef float v2f __attribute__((ext_vector_type(2)));
typedef float v8f __attribute__((ext_vector_type(8)));

// Volume: [B=2, C=1, H=192, W=192, D=192], D contiguous.
constexpr int SZ = 192;
constexpr int TW = 16;    // w outputs per WG
constexpr int TD = 32;    // d outputs per WG
constexpr int TH = 24;    // h outputs per WG
constexpr int NT = 128;   // threads per WG (4 wave32)
constexpr int GW = SZ / TW;           // 12
constexpr int GD = SZ / TD;           // 6
constexpr int GH = SZ / TH;           // 8
constexpr int NBLK = 2 * GW * GD * GH; // 1152
constexpr int RS = 36;                // LDS row stride (34 used + pad)
constexpr float KVOL_INV = 1.0f / 27.0f;
constexpr float NTOT = 2.0f * 192.0f * 192.0f * 192.0f; // 14155776

__global__ __launch_bounds__(NT)
void lncc_main(const float* __restrict__ pred,
               const float* __restrict__ target,
               float* __restrict__ partials) {
  __shared__ float rawI[18 * RS];
  __shared__ float rawJ[18 * RS];
  __shared__ float Pq[5 * 18 * RS];        // d-filtered I,J,II,JJ,IJ (w-halo rows 0..17)
  __shared__ float ring[3 * 5 * 16 * RS];  // (w,d)-filtered plane sums, 3-deep h-ring
  __shared__ float red[NT];

  const int tid = threadIdx.x;
  int id = blockIdx.x;
  const int ht = id % GH; id /= GH;
  const int dt = id % GD; id /= GD;
  const int wt = id % GW; id /= GW;
  const int b  = id;
  const int h0 = ht * TH, d0 = dt * TD, w0 = wt * TW;

  const int wv   = tid >> 5;    // wave id 0..3
  const int lane = tid & 31;
  const int mrow = lane & 15;   // M for A, N for B/C
  const int half = lane >> 4;   // 0 / 1

  float acc = 0.0f;

  for (int hp = h0 - 1; hp <= h0 + TH; ++hp) {
    const int slot = (hp + 3) % 3;  // hp >= -1

    if (hp >= 0 && hp < SZ) {
      // ---- stage 1: coalesced load of 18x34 halo plane (both arrays) ----
      for (int i = tid; i < 18 * 34; i += NT) {
        const int r = i / 34, c = i % 34;
        const int w = w0 - 1 + r;
        const int d = d0 - 1 + c;
        float vi = 0.0f, vj = 0.0f;
        if ((unsigned)w < (unsigned)SZ && (unsigned)d < (unsigned)SZ) {
          const int off = ((b * SZ + hp) * SZ + w) * SZ + d;
          vi = pred[off];
          vj = target[off];
          if (hp + 1 < SZ) {  // prefetch next h-plane (global_prefetch_b8)
            __builtin_prefetch(pred + off + SZ * SZ, 0, 0);
            __builtin_prefetch(target + off + SZ * SZ, 0, 0);
          }
        }
        rawI[r * RS + c] = vi;
        rawJ[r * RS + c] = vj;
      }
      __syncthreads();

      // ---- stage 2: 3-tap sum along contiguous d of the 5 quantities ----
      for (int i = tid; i < 18 * TD; i += NT) {
        const int r = i / TD, c = i % TD;
        const int base = r * RS + c;  // raw col c == d0-1+c; center = c+1
        const float i0 = rawI[base], i1 = rawI[base + 1], i2 = rawI[base + 2];
        const float j0 = rawJ[base], j1 = rawJ[base + 1], j2 = rawJ[base + 2];
        Pq[(0 * 18 + r) * RS + c] = i0 + i1 + i2;
        Pq[(1 * 18 + r) * RS + c] = j0 + j1 + j2;
        Pq[(2 * 18 + r) * RS + c] = i0 * i0 + i1 * i1 + i2 * i2;
        Pq[(3 * 18 + r) * RS + c] = j0 * j0 + j1 * j1 + j2 * j2;
        Pq[(4 * 18 + r) * RS + c] = i0 * j0 + i1 * j1 + i2 * j2;
      }
      __syncthreads();

      // ---- stage 3: w-direction 3-tap via banded f32 WMMA (exact 0/1 weights)
      // U[j,d] = sum_{w'=0..17} band(j,w') * Pq[w'][d]; core w'=1..16 via 4x K=4
      // WMMA, halo rows w'=0,17 folded into C rows M=0 / M=15 with VALU.
      for (int job = wv; job < 10; job += 4) {   // 5 quantities x 2 d-tiles
        const int q = job >> 1;
        const int dbase = (job & 1) * 16;
        v8f cacc = {0.f, 0.f, 0.f, 0.f, 0.f, 0.f, 0.f, 0.f};
#pragma unroll
        for (int ch = 0; ch < 4; ++ch) {
          // A 16x4 layout: VGPR r, lanes0-15 -> K=r; lanes16-31 -> K=r+2
          const int k0 = ch * 4 + 2 * half;
          const int k1 = k0 + 1;
          const int e0 = k0 - mrow, e1 = k1 - mrow;
          v2f a, bb;
          a.x = (e0 >= -1 && e0 <= 1) ? 1.0f : 0.0f;  // tridiagonal band
          a.y = (e1 >= -1 && e1 <= 1) ? 1.0f : 0.0f;
          // B 4x16 layout mirrors A: VGPR r holds row K=r (lanes0-15) / K=r+2
          bb.x = Pq[(q * 18 + 1 + k0) * RS + dbase + mrow];
          bb.y = Pq[(q * 18 + 1 + k1) * RS + dbase + mrow];
          cacc = __builtin_amdgcn_wmma_f32_16x16x4_f32(
              false, a, false, bb, (short)0, cacc, false, false);
        }
        // halo fixups: row M=0 lives in c[0] lanes0-15; M=15 in c[7] lanes16-31
        const float f0  = (half == 0) ? Pq[(q * 18 + 0)  * RS + dbase + mrow] : 0.0f;
        const float f15 = (half == 1) ? Pq[(q * 18 + 17) * RS + dbase + mrow] : 0.0f;
        cacc[0] += f0;
        cacc[7] += f15;
        // scatter C/D to ring: VGPR r -> rows M=r (lanes0-15), M=r+8 (lanes16-31)
#pragma unroll
        for (int r = 0; r < 8; ++r) {
          const int M = r + 8 * half;
          ring[((slot * 5 + q) * 16 + M) * RS + dbase + mrow] = cacc[r];
        }
      }
    } else {
      // plane outside the volume -> zero ring slot (zero padding)
      for (int i = tid; i < 5 * 16 * RS; i += NT)
        ring[slot * 5 * 16 * RS + i] = 0.0f;
    }
    __syncthreads();

    // ---- stage 4: h-direction 3-tap (ring sum) + fused ZNCC ----
    const int hout = hp - 1;
    if (hout >= h0 && hout < h0 + TH) {
      for (int i = tid; i < TW * TD; i += NT) {
        const int r = i / TD, c = i % TD;
        float SI = 0.f, SJ = 0.f, SII = 0.f, SJJ = 0.f, SIJ = 0.f;
#pragma unroll
        for (int s = 0; s < 3; ++s) {
          const int base = (s * 5 * 16 + r) * RS + c;
          SI  += ring[base];
          SJ  += ring[base + 1 * 16 * RS];
          SII += ring[base + 2 * 16 * RS];
          SJJ += ring[base + 3 * 16 * RS];
          SIJ += ring[base + 4 * 16 * RS];
        }
        const float cross = SIJ - SI * SJ * KVOL_INV;
        const float pvar  = SII - SI * SI * KVOL_INV;
        const float tvar  = SJJ - SJ * SJ * KVOL_INV;
        acc += (cross * cross) / (tvar * pvar + 1e-5f);
      }
    }
    __syncthreads();
  }

  // deterministic in-WG tree reduction
  red[tid] = acc;
  __syncthreads();
  for (int s = NT / 2; s > 0; s >>= 1) {
    if (tid < s) red[tid] += red[tid + s];
    __syncthreads();
  }
  if (tid == 0) partials[blockIdx.x] = red[0];
}

__global__ __launch_bounds__(256)
void lncc_reduce(const float* __restrict__ partials, float* __restrict__ out) {
  __shared__ float red[256];
  float s = 0.0f;
  for (int i = threadIdx.x; i < NBLK; i += 256) s += partials[i];  // fixed order
  red[threadIdx.x] = s;
  __syncthreads();
  for (int k = 128; k > 0; k >>= 1) {
    if (threadIdx.x < k) red[threadIdx.x] += red[threadIdx.x + k];
    __syncthreads();
  }
  if (threadIdx.x == 0) out[0] = -red[0] / NTOT;
}

extern "C" void kernel_launch(void* const* d_in, const int* in_sizes, int n_in,
                              void* d_out, int out_size, void* d_ws, size_t ws_size,
                              hipStream_t stream) {
  (void)in_sizes; (void)n_in; (void)out_size; (void)ws_size;
  const float* pred   = (const float*)d_in[0];
  const float* target = (const float*)d_in[1];
  float* out      = (float*)d_out;
  float* partials = (float*)d_ws;  // NBLK floats of scratch
  lncc_main<<<NBLK, NT, 0, stream>>>(pred, target, partials);
  lncc_reduce<<<1, 256, 0, stream>>>(partials, out);
}